// ContrastiveEncoder_45835890983033
// MI455X (gfx1250) — compile-verified
//
#include <hip/hip_runtime.h>

// GCN: x -> GCNConv(1->64) relu -> GCNConv(64->64) relu -> concat -> mean-pool -> linear(128->10)
// N=100000, E=3200000, G=512, HID=64, OUT=10

#define HID 64
#define NGRAPH 512
#define OUTC 10

typedef __attribute__((ext_vector_type(2))) float v2f;
typedef __attribute__((ext_vector_type(8))) float v8f;

static __device__ __forceinline__ void atomAddF(float* p, float v) {
  unsafeAtomicAdd(p, v);  // hardware global_atomic_add_f32
}

// ---- ws init: deg=1.0 (self loop), everything else 0 ----
__global__ void k_init(float* __restrict__ ws, int nNodes, int total) {
  int t = blockIdx.x * blockDim.x + threadIdx.x;
  if (t >= total) return;
  ws[t] = (t < nNodes) ? 1.0f : 0.0f;
}

// ---- in-degree via scalar atomics ----
__global__ void k_degree(const long long* __restrict__ dst, float* __restrict__ deg, int nEdges) {
  int t = blockIdx.x * blockDim.x + threadIdx.x;
  if (t >= nEdges) return;
  atomAddF(deg + (int)dst[t], 1.0f);
}

// ---- dinv = rsqrt(max(deg,1)) ----
__global__ void k_dinv(const float* __restrict__ deg, float* __restrict__ dinv, int nNodes) {
  int t = blockIdx.x * blockDim.x + threadIdx.x;
  if (t >= nNodes) return;
  dinv[t] = __frsqrt_rn(fmaxf(deg[t], 1.0f));
}

// ---- layer-1 scalar scatter: s1[d] += x[s]*dinv[s]*dinv[d] ----
__global__ void k_scatter1(const long long* __restrict__ src, const long long* __restrict__ dst,
                           const float* __restrict__ x, const float* __restrict__ dinv,
                           float* __restrict__ s1, int nEdges) {
  int t = blockIdx.x * blockDim.x + threadIdx.x;
  if (t >= nEdges) return;
  int s = (int)src[t], d = (int)dst[t];
  atomAddF(s1 + d, x[s] * dinv[s] * dinv[d]);
}

// ---- layer-1 epilogue (rank-1 expansion): x1[n,f] = relu((s1[n]+x[n]*dinv[n]^2)*W1[f]+b1[f]) ----
__global__ void k_layer1(const float* __restrict__ s1, const float* __restrict__ x,
                         const float* __restrict__ dinv, const float* __restrict__ W1,
                         const float* __restrict__ b1, float* __restrict__ x1, int nNodes) {
  int t = blockIdx.x * blockDim.x + threadIdx.x;
  if (t >= nNodes * HID) return;
  int n = t >> 6, f = t & 63;
  float di = dinv[n];
  float agg = s1[n] + x[n] * di * di;
  x1[t] = fmaxf(agg * W1[f] + b1[f], 0.0f);
}

// ---- dense GEMM g = x1 @ W2 (N x 64 @ 64 x 64), f32 WMMA 16x16x4, 1 node-tile per wave ----
__global__ void k_gemm_wmma(const float* __restrict__ x1, const float* __restrict__ W2,
                            float* __restrict__ g, int nNodes, int nTiles) {
  int wave = blockIdx.x * (blockDim.x >> 5) + (threadIdx.x >> 5);
  if (wave >= nTiles) return;
  int lane = threadIdx.x & 31;
  int lo = lane & 15;        // row/col within 16-half
  int hi = lane >> 4;        // 0: K={0,1}, 1: K={2,3} per ISA 16x4 f32 A layout
  int m0 = wave * 16;
  bool fullTile = (m0 + 16 <= nNodes);   // wave-uniform

  // A-row (clamped only for the tail tile)
  int arow = m0 + lo;
  if (!fullTile && arow >= nNodes) arow = nNodes - 1;

  // Preload all 16 A fragments (K=0..63) for this 16-row tile: 32 VGPRs,
  // immediate-offset b64 loads off one base address.
  const float* ap = x1 + (long long)arow * HID + 2 * hi;
  v2f afrag[16];
#pragma unroll
  for (int k = 0; k < 16; ++k)
    afrag[k] = *(const v2f*)(ap + 4 * k);

  // Per-wave store base: row = m0 + 8*hi + r, col = n0 + lo  ->  all offsets immediate
  float* gbase = g + ((long long)(m0 + 8 * hi) * HID + lo);
  const float* wbase = W2 + (2 * hi) * HID + lo;

#pragma unroll
  for (int nt = 0; nt < 4; ++nt) {
    int n0 = nt * 16;
    v8f acc = {};
#pragma unroll
    for (int k = 0; k < 16; ++k) {
      v2f b;
      b.x = wbase[(4 * k) * HID + n0];       // W2[4k+2hi,   n0+lo]
      b.y = wbase[(4 * k + 1) * HID + n0];   // W2[4k+2hi+1, n0+lo]
      // 8 args: (neg_a, A, neg_b, B, c_mod, C, reuse_a, reuse_b)
      acc = __builtin_amdgcn_wmma_f32_16x16x4_f32(false, afrag[k], false, b,
                                                  (short)0, acc, false, false);
    }
    if (fullTile) {
      // Branch-free immediate-offset stores (common case; N%16==0 -> always)
#pragma unroll
      for (int r = 0; r < 8; ++r)
        gbase[r * HID + n0] = acc[r];
    } else {
#pragma unroll
      for (int r = 0; r < 8; ++r) {
        int row = m0 + r + 8 * hi;
        if (row < nNodes) gbase[r * HID + n0] = acc[r];
      }
    }
  }
}

// ---- layer-2 edge scatter: warp per edge, 2 features per lane; L2-resident gather/atomic ----
__global__ void k_scatter2(const long long* __restrict__ src, const long long* __restrict__ dst,
                           const float* __restrict__ dinv, const float* __restrict__ g,
                           float* __restrict__ h2, int nEdges) {
  int warp = blockIdx.x * (blockDim.x >> 5) + (threadIdx.x >> 5);
  if (warp >= nEdges) return;
  int lane = threadIdx.x & 31;
  int s = (int)src[warp], d = (int)dst[warp];
  float nrm = dinv[s] * dinv[d];
  int f = lane * 2;
  v2f v = *(const v2f*)(g + (long long)s * HID + f);
  float* o = h2 + (long long)d * HID + f;
  atomAddF(o, v.x * nrm);
  atomAddF(o + 1, v.y * nrm);
}

// ---- layer-2 epilogue fused with mean-pool accumulation ----
__global__ void k_finish_pool(const float* __restrict__ g, const float* __restrict__ h2,
                              const float* __restrict__ x1, const float* __restrict__ dinv,
                              const float* __restrict__ b2, const long long* __restrict__ batch,
                              float* __restrict__ pooled, int nNodes) {
  int t = blockIdx.x * blockDim.x + threadIdx.x;
  if (t >= nNodes * HID) return;
  int n = t >> 6, f = t & 63;
  float di = dinv[n];
  float x2 = fmaxf(h2[t] + g[t] * di * di + b2[f], 0.0f);
  int b = (int)batch[n];
  atomAddF(pooled + b * (2 * HID) + f, x1[t]);       // first half: x1
  atomAddF(pooled + b * (2 * HID) + HID + f, x2);    // second half: x2
}

// ---- per-graph node counts ----
__global__ void k_count(const long long* __restrict__ batch, float* __restrict__ cnt, int nNodes) {
  int t = blockIdx.x * blockDim.x + threadIdx.x;
  if (t >= nNodes) return;
  atomAddF(cnt + (int)batch[t], 1.0f);
}

// ---- readout: out[g,o] = (pooled[g,:]/max(cnt,1)) @ Wl + bl ----
__global__ void k_readout(const float* __restrict__ pooled, const float* __restrict__ cnt,
                          const float* __restrict__ Wl, const float* __restrict__ bl,
                          float* __restrict__ out) {
  int t = blockIdx.x * blockDim.x + threadIdx.x;
  if (t >= NGRAPH * OUTC) return;
  int gr = t / OUTC, o = t % OUTC;
  float inv = 1.0f / fmaxf(cnt[gr], 1.0f);
  float acc = bl[o];
  const float* p = pooled + gr * (2 * HID);
#pragma unroll 8
  for (int k = 0; k < 2 * HID; ++k)
    acc = fmaf(p[k] * inv, Wl[k * OUTC + o], acc);
  out[t] = acc;
}

extern "C" void kernel_launch(void* const* d_in, const int* in_sizes, int n_in,
                              void* d_out, int out_size, void* d_ws, size_t ws_size,
                              hipStream_t stream) {
  const float*     x     = (const float*)d_in[0];
  const long long* ei    = (const long long*)d_in[1];
  const long long* batch = (const long long*)d_in[2];
  const float*     W1    = (const float*)d_in[3];
  const float*     b1    = (const float*)d_in[4];
  const float*     W2    = (const float*)d_in[5];
  const float*     b2    = (const float*)d_in[6];
  const float*     Wl    = (const float*)d_in[7];
  const float*     bl    = (const float*)d_in[8];
  float* out = (float*)d_out;

  const int N = in_sizes[0];
  const int E = in_sizes[1] / 2;
  const long long* srcIdx = ei;
  const long long* dstIdx = ei + E;

  // Workspace layout (floats)
  float* ws     = (float*)d_ws;
  float* deg    = ws;                       // N
  float* dinv   = deg + N;                  // N
  float* s1     = dinv + N;                 // N
  float* x1     = s1 + N;                   // N*64
  float* g      = x1 + (long long)N * HID;  // N*64
  float* h2     = g + (long long)N * HID;   // N*64
  float* pooled = h2 + (long long)N * HID;  // 512*128
  float* cnt    = pooled + NGRAPH * 2 * HID;// 512
  const int totalInit = 3 * N + 3 * N * HID + NGRAPH * 2 * HID + NGRAPH;

  const int B = 256;
  k_init<<<(totalInit + B - 1) / B, B, 0, stream>>>(ws, N, totalInit);
  k_degree<<<(E + B - 1) / B, B, 0, stream>>>(dstIdx, deg, E);
  k_dinv<<<(N + B - 1) / B, B, 0, stream>>>(deg, dinv, N);
  k_scatter1<<<(E + B - 1) / B, B, 0, stream>>>(srcIdx, dstIdx, x, dinv, s1, E);
  k_layer1<<<(N * HID + B - 1) / B, B, 0, stream>>>(s1, x, dinv, W1, b1, x1, N);

  int nTiles = (N + 15) / 16;
  int wavesPerBlk = B / 32;  // 8 waves -> 8 node tiles per block
  k_gemm_wmma<<<(nTiles + wavesPerBlk - 1) / wavesPerBlk, B, 0, stream>>>(x1, W2, g, N, nTiles);

  int warpsPerBlk = B / 32;
  k_scatter2<<<(E + warpsPerBlk - 1) / warpsPerBlk, B, 0, stream>>>(srcIdx, dstIdx, dinv, g, h2, E);
  k_finish_pool<<<(N * HID + B - 1) / B, B, 0, stream>>>(g, h2, x1, dinv, b2, batch, pooled, N);
  k_count<<<(N + B - 1) / B, B, 0, stream>>>(batch, cnt, N);
  k_readout<<<(NGRAPH * OUTC + B - 1) / B, B, 0, stream>>>(pooled, cnt, Wl, bl, out);
}